// RoiPooling_27590869909528
// MI455X (gfx1250) — compile-verified
//
#include <hip/hip_runtime.h>

// ROI pooling (crop + legacy-TF bilinear resize to 14x14) for MI455X / gfx1250.
//
// Shapes (from reference setup_inputs):
//   feature_map: [8, 64, 64, 256] f32   (33.5 MB -> fully L2-resident, 192 MB L2)
//   rois:        [8, 300, 4]      f32
//   out:         [8, 300, 14, 14, 256] f32 (482 MB -> the bandwidth bottleneck)
//
// Design: store-BW-bound kernel. b128 loads for the 4 corner gathers (L2 hits),
// non-temporal b128 stores for the output stream so it does not evict the
// feature map from L2. One 256-thread block (8 wave32s) per ROI.

typedef float v4f __attribute__((ext_vector_type(4)));

namespace {
constexpr int kB = 8;
constexpr int kH = 64;
constexpr int kW = 64;
constexpr int kC = 256;
constexpr int kR = 300;
constexpr int kP = 14;
constexpr int kC4 = kC / 4;  // 64 float4 channel groups per pixel
}  // namespace

__global__ __launch_bounds__(256) void roi_pool_bilinear(
    const float* __restrict__ fmap,
    const float* __restrict__ rois,
    float* __restrict__ out) {
  const int blk = blockIdx.x;          // b * kR + r
  const int b   = blk / kR;

  // ---- ROI parameters (uniform across the block) -------------------------
  // Semantics must bit-match the reference:
  //   x0 = (int)roi.x (truncate, roi >= 0 so == floor)
  //   w  = (int)max(1.0f, roi.z - roi.x)
  //   scale = (float)w / 14.0f  (true division, same rounding as jnp h/P)
  const v4f roi = ((const v4f*)rois)[blk];
  const int   x0 = (int)roi[0];
  const int   y0 = (int)roi[1];
  const int   w  = (int)fmaxf(1.0f, roi[2] - roi[0]);
  const int   h  = (int)fmaxf(1.0f, roi[3] - roi[1]);
  const float sx = (float)w / (float)kP;
  const float sy = (float)h / (float)kP;

  const int tid = threadIdx.x;
  const int cg  = tid & (kC4 - 1);     // channel float4 group: 0..63
  const int pg  = tid >> 6;            // pixel within iteration: 0..3

  const v4f* __restrict__ fm4 =
      (const v4f*)fmap + (size_t)b * (size_t)(kH * kW * kC4);
  v4f* __restrict__ o4 =
      (v4f*)out + (size_t)blk * (size_t)(kP * kP * kC4);

  // 196 pixels, 4 per iteration, 49 iterations.
#pragma unroll 2
  for (int it = 0; it < (kP * kP) / 4; ++it) {
    const int pix = it * 4 + pg;
    const int py  = pix / kP;
    const int px  = pix - py * kP;

    // Legacy TF resize: src = dst * (in/out), floor for top-left corner.
    const float iy = (float)py * sy;
    const int   ty = (int)iy;                 // iy >= 0 -> trunc == floor
    const float fy = iy - (float)ty;
    const int   by = min(ty + 1, h - 1);

    const float ix = (float)px * sx;
    const int   lx = (int)ix;
    const float fx = ix - (float)lx;
    const int   rx = min(lx + 1, w - 1);

    const int gyt = min(max(y0 + ty, 0), kH - 1);
    const int gyb = min(max(y0 + by, 0), kH - 1);
    const int gxl = min(max(x0 + lx, 0), kW - 1);
    const int gxr = min(max(x0 + rx, 0), kW - 1);

    // 4 corner gathers: each is 1 KB contiguous across the 64 cg lanes
    // (b128 per lane) and an L2 hit after the first touch of the image.
    const int rowT = gyt * kW;
    const int rowB = gyb * kW;
    const v4f tl = fm4[(rowT + gxl) * kC4 + cg];
    const v4f tr = fm4[(rowT + gxr) * kC4 + cg];
    const v4f bl = fm4[(rowB + gxl) * kC4 + cg];
    const v4f br = fm4[(rowB + gxr) * kC4 + cg];

    // Same algebraic form as the reference.
    const v4f top = tl + (tr - tl) * fx;
    const v4f bot = bl + (br - bl) * fx;
    const v4f res = top + (bot - top) * fy;

    // Non-temporal b128 store (th:TH_STORE_NT): the 482 MB output stream
    // must not evict the L2-resident feature map.
    __builtin_nontemporal_store(res, &o4[pix * kC4 + cg]);
  }
}

extern "C" void kernel_launch(void* const* d_in, const int* in_sizes, int n_in,
                              void* d_out, int out_size, void* d_ws,
                              size_t ws_size, hipStream_t stream) {
  (void)in_sizes; (void)n_in; (void)d_ws; (void)ws_size; (void)out_size;
  const float* fmap = (const float*)d_in[0];
  const float* rois = (const float*)d_in[1];
  float*       out  = (float*)d_out;

  dim3 grid(kB * kR);   // 2400 blocks, one per ROI
  dim3 block(256);      // 8 wave32s
  hipLaunchKernelGGL(roi_pool_bilinear, grid, block, 0, stream,
                     fmap, rois, out);
}